// st_gcn_80066780332498
// MI455X (gfx1250) — compile-verified
//
#include <hip/hip_runtime.h>

// ST-GCN block for MI455X (gfx1250): fp32 WMMA 16x16x4 everywhere.
// Memory-bound problem (~335MB mandatory traffic @ 23.3TB/s dominates the
// ~95 GFLOP of math), so we keep full fp32 precision via V_WMMA_F32_16X16X4_F32
// and use the CDNA5 async global->LDS engine (ASYNCcnt) for panel staging.

typedef float v2f __attribute__((ext_vector_type(2)));
typedef float v8f __attribute__((ext_vector_type(8)));

#define NN 256
#define CCH 256
#define TTT 4
#define VVV 128
#define EPSBN 1e-5f

// ---- CDNA5 async global->LDS staging (guarded: falls back to plain loads) --
#if defined(__gfx1250__) && \
    __has_builtin(__builtin_amdgcn_global_load_async_to_lds_b128) && \
    __has_builtin(__builtin_amdgcn_s_wait_asynccnt)
#define USE_ASYNC_LDS 1
#else
#define USE_ASYNC_LDS 0
#endif

#if USE_ASYNC_LDS
#define AS1 __attribute__((address_space(1)))
#define AS3 __attribute__((address_space(3)))
typedef int vi4 __attribute__((vector_size(16)));  // matches builtin's V4i
// Per-lane copy of 16 contiguous bytes global -> LDS, tracked by ASYNCcnt.
// LDS flat address keeps its offset in addr[31:0] (ISA 10.2), so truncating
// the generic pointer yields the as(3) offset (standard CK idiom).
static __device__ __forceinline__ void async_cp16(void* lds_dst,
                                                  const void* gsrc) {
  __builtin_amdgcn_global_load_async_to_lds_b128(
      (AS1 vi4*)(uintptr_t)gsrc,
      (AS3 vi4*)(uint32_t)(uintptr_t)lds_dst,
      /*offset=*/0, /*cpol=*/0);
}
static __device__ __forceinline__ void async_wait0() {
  __builtin_amdgcn_s_wait_asynccnt(0);
}
#endif

static __device__ __forceinline__ v8f wmma_k4(v2f a, v2f b, v8f c) {
  // (neg_a, A, neg_b, B, c_mod, C, reuse_a, reuse_b)
  return __builtin_amdgcn_wmma_f32_16x16x4_f32(false, a, false, b, (short)0, c,
                                               false, false);
}

static __device__ __forceinline__ v8f v8zero() {
  v8f z;
#pragma unroll
  for (int i = 0; i < 8; ++i) z[i] = 0.0f;
  return z;
}

// -------------------------------------------------------------------------
// K1: xa[n,:,t,:] = X[n,:,t,:] (256x128) @ A[n,t] (128x128)
// grid (N*T, C/64), 256 threads (8 waves). Wave -> 16 rows x 64 cols.
// -------------------------------------------------------------------------
__global__ __launch_bounds__(256) void k1_xa(const float* __restrict__ X,
                                             const float* __restrict__ A,
                                             float* __restrict__ XA) {
  const int nt = blockIdx.x;
  const int n = nt >> 2, t = nt & 3;
  const int mBase = blockIdx.y * 64;
  const int tid = threadIdx.x;
  const int wave = tid >> 5, lane = tid & 31;
  const int mTile = wave >> 1;      // 0..3 (16 rows each)
  const int nGroup = wave & 1;      // 0..1 (4 col-tiles each)
  const int l16 = lane & 15;
  const int klo = (lane >> 4) << 1; // 0 or 2 (A/B half-wave K split)

  __shared__ float Xs[64][36];   // [m][k] padded: conflict-free frag-A reads
  __shared__ float As[32][144];  // [k][w] padded: conflict-free frag-B reads

  v8f acc[4];
#pragma unroll
  for (int j = 0; j < 4; ++j) acc[j] = v8zero();

  const float* Xrow = X + ((size_t)(n * CCH + mBase) * TTT + t) * VVV;
  const float* Ablk = A + (size_t)nt * VVV * VVV;

  for (int k0 = 0; k0 < VVV; k0 += 32) {
#if USE_ASYNC_LDS
    // Xs: 64 rows x 32 floats = 512 x 16B chunks (2 per thread)
#pragma unroll
    for (int j = 0; j < 2; ++j) {
      int i = tid + j * 256;
      int m = i >> 3, q = (i & 7) * 4;
      async_cp16(&Xs[m][q], Xrow + (size_t)m * (TTT * VVV) + k0 + q);
    }
    // As: 32 rows x 128 floats = 1024 x 16B chunks (4 per thread)
#pragma unroll
    for (int j = 0; j < 4; ++j) {
      int i = tid + j * 256;
      int k = i >> 5, q = (i & 31) * 4;
      async_cp16(&As[k][q], Ablk + (size_t)(k0 + k) * VVV + q);
    }
#else
    for (int i = tid; i < 64 * 32; i += 256) {
      int m = i >> 5, k = i & 31;
      Xs[m][k] = Xrow[(size_t)m * (TTT * VVV) + (k0 + k)];
    }
    for (int i = tid; i < 32 * 128; i += 256) {
      int k = i >> 7, w = i & 127;
      As[k][w] = Ablk[(size_t)(k0 + k) * VVV + w];
    }
#endif
    if (k0 + 32 < VVV) {
      // CDNA5 global_prefetch_b8 of next A chunk
      __builtin_prefetch(&Ablk[(size_t)(k0 + 32 + (tid & 31)) * VVV], 0, 1);
    }
#if USE_ASYNC_LDS
    async_wait0();
#endif
    __syncthreads();
#pragma unroll
    for (int kk = 0; kk < 32; kk += 4) {
      v2f a;
      a.x = Xs[mTile * 16 + l16][kk + klo];
      a.y = Xs[mTile * 16 + l16][kk + klo + 1];
#pragma unroll
      for (int j = 0; j < 4; ++j) {
        const int col = (nGroup * 4 + j) * 16 + l16;
        v2f b;
        b.x = As[kk + klo][col];
        b.y = As[kk + klo + 1][col];
        acc[j] = wmma_k4(a, b, acc[j]);
      }
    }
    __syncthreads();
  }

  const int mOff = (lane < 16) ? 0 : 8;
#pragma unroll
  for (int j = 0; j < 4; ++j) {
    const int w = (nGroup * 4 + j) * 16 + l16;
#pragma unroll
    for (int i = 0; i < 8; ++i) {
      const int m = mBase + mTile * 16 + mOff + i;
      XA[((size_t)(n * CCH + m) * TTT + t) * VVV + w] = acc[j][i];
    }
  }
}

// -------------------------------------------------------------------------
// K2: H = relu(W0[t]^T @ xa) + relu(W1[t]^T @ xa) + X   per (n,t)
// A-matrix = W^T (h rows, c cols); W stored [t][c][h] (h contiguous).
// -------------------------------------------------------------------------
__global__ __launch_bounds__(256) void k2_gcn(const float* __restrict__ XA,
                                              const float* __restrict__ W0,
                                              const float* __restrict__ W1,
                                              const float* __restrict__ X,
                                              float* __restrict__ H) {
  const int nt = blockIdx.x;
  const int n = nt >> 2, t = nt & 3;
  const int hBase = blockIdx.y * 64;
  const int tid = threadIdx.x;
  const int wave = tid >> 5, lane = tid & 31;
  const int mTile = wave >> 1;
  const int nGroup = wave & 1;
  const int l16 = lane & 15;
  const int klo = (lane >> 4) << 1;

  __shared__ float W0s[32][72];  // [k][h]
  __shared__ float W1s[32][72];
  __shared__ float Bs[32][144];  // [k][w]

  v8f acc0[4], acc1[4];
#pragma unroll
  for (int j = 0; j < 4; ++j) { acc0[j] = v8zero(); acc1[j] = v8zero(); }

  const float* w0t = W0 + (size_t)t * CCH * CCH;
  const float* w1t = W1 + (size_t)t * CCH * CCH;

  for (int k0 = 0; k0 < CCH; k0 += 32) {
#if USE_ASYNC_LDS
    // W panels: 64 x 32 floats = 512 x 16B chunks each (2 per thread)
#pragma unroll
    for (int j = 0; j < 2; ++j) {
      int i = tid + j * 256;
      int k = i >> 3, q = (i & 7) * 4;
      async_cp16(&W0s[k][q], w0t + (size_t)(k0 + k) * CCH + hBase + q);
      async_cp16(&W1s[k][q], w1t + (size_t)(k0 + k) * CCH + hBase + q);
    }
    // xa panel: 32 x 128 floats = 1024 x 16B chunks (4 per thread)
#pragma unroll
    for (int j = 0; j < 4; ++j) {
      int i = tid + j * 256;
      int k = i >> 5, q = (i & 31) * 4;
      async_cp16(&Bs[k][q],
                 XA + ((size_t)(n * CCH + (k0 + k)) * TTT + t) * VVV + q);
    }
    async_wait0();
#else
    for (int i = tid; i < 32 * 64; i += 256) {
      int k = i >> 6, m = i & 63;
      W0s[k][m] = w0t[(size_t)(k0 + k) * CCH + hBase + m];
      W1s[k][m] = w1t[(size_t)(k0 + k) * CCH + hBase + m];
    }
    for (int i = tid; i < 32 * 128; i += 256) {
      int k = i >> 7, w = i & 127;
      Bs[k][w] = XA[((size_t)(n * CCH + (k0 + k)) * TTT + t) * VVV + w];
    }
#endif
    __syncthreads();
#pragma unroll
    for (int kk = 0; kk < 32; kk += 4) {
      const int mLoc = mTile * 16 + l16;
      v2f a0, a1;
      a0.x = W0s[kk + klo][mLoc];  a0.y = W0s[kk + klo + 1][mLoc];
      a1.x = W1s[kk + klo][mLoc];  a1.y = W1s[kk + klo + 1][mLoc];
#pragma unroll
      for (int j = 0; j < 4; ++j) {
        const int col = (nGroup * 4 + j) * 16 + l16;
        v2f b;
        b.x = Bs[kk + klo][col];
        b.y = Bs[kk + klo + 1][col];
        acc0[j] = wmma_k4(a0, b, acc0[j]);
        acc1[j] = wmma_k4(a1, b, acc1[j]);
      }
    }
    __syncthreads();
  }

  const int mOff = (lane < 16) ? 0 : 8;
#pragma unroll
  for (int j = 0; j < 4; ++j) {
    const int w = (nGroup * 4 + j) * 16 + l16;
#pragma unroll
    for (int i = 0; i < 8; ++i) {
      const int h = hBase + mTile * 16 + mOff + i;
      const size_t idx = ((size_t)(n * CCH + h) * TTT + t) * VVV + w;
      float r0 = acc0[j][i]; r0 = r0 > 0.0f ? r0 : 0.0f;
      float r1 = acc1[j][i]; r1 = r1 > 0.0f ? r1 : 0.0f;
      H[idx] = r0 + r1 + X[idx];
    }
  }
}

// -------------------------------------------------------------------------
// BN train-mode stats over (N,T,V) per channel -> affine scale/shift.
// One block per channel.
// -------------------------------------------------------------------------
__global__ __launch_bounds__(256) void k_bnstats(const float* __restrict__ src,
                                                 const float* __restrict__ g,
                                                 const float* __restrict__ b,
                                                 float* __restrict__ scale,
                                                 float* __restrict__ shift) {
  const int c = blockIdx.x;
  const int tid = threadIdx.x;
  float s1 = 0.0f, s2 = 0.0f;
  for (int n = 0; n < NN; ++n) {
    const float* p = src + (size_t)(n * CCH + c) * TTT * VVV;
    for (int i = tid; i < TTT * VVV; i += 256) {
      float v = p[i];
      s1 += v;
      s2 += v * v;
    }
  }
  __shared__ float r1[256], r2[256];
  r1[tid] = s1; r2[tid] = s2;
  __syncthreads();
  for (int s = 128; s > 0; s >>= 1) {
    if (tid < s) { r1[tid] += r1[tid + s]; r2[tid] += r2[tid + s]; }
    __syncthreads();
  }
  if (tid == 0) {
    const float cnt = (float)NN * (float)(TTT * VVV);
    const float mean = r1[0] / cnt;
    const float var = r2[0] / cnt - mean * mean;  // biased variance
    const float inv = rsqrtf(var + EPSBN) * g[c];
    scale[c] = inv;
    shift[c] = b[c] - mean * inv;
  }
}

// -------------------------------------------------------------------------
// K4: conv(3,1) pad(1,0) as 3 shifted WMMA GEMMs; BN1+PReLU applied while
// staging H tiles. out2[n,h,t,:] = bias[h] + sum_k Wk @ prelu(bn1(H[..t+k-1..]))
// conv_w layout (Cout, Cin, 3, 1): idx = (h*C + c)*3 + k
// (plain staging: weight gather is stride-3, B panel needs a transform)
// -------------------------------------------------------------------------
__global__ __launch_bounds__(256) void k4_conv(const float* __restrict__ Hbuf,
                                               const float* __restrict__ CW,
                                               const float* __restrict__ CB,
                                               const float* __restrict__ sc1,
                                               const float* __restrict__ sh1,
                                               const float* __restrict__ aTcn,
                                               float* __restrict__ OUT2) {
  const int nt = blockIdx.x;
  const int n = nt >> 2, t = nt & 3;
  const int hBase = blockIdx.y * 64;
  const int tid = threadIdx.x;
  const int wave = tid >> 5, lane = tid & 31;
  const int mTile = wave >> 1;
  const int nGroup = wave & 1;
  const int l16 = lane & 15;
  const int klo = (lane >> 4) << 1;
  const float at = aTcn[0];

  __shared__ float Ws[32][72];   // [c][h]
  __shared__ float Bs[32][144];  // [c][w]

  v8f acc[4];
#pragma unroll
  for (int j = 0; j < 4; ++j) acc[j] = v8zero();

  for (int k = 0; k < 3; ++k) {
    const int tt = t + k - 1;
    if (tt < 0 || tt >= TTT) continue;  // uniform across block (t from blockIdx)
    for (int c0 = 0; c0 < CCH; c0 += 32) {
      for (int i = tid; i < 32 * 64; i += 256) {
        int cc = i >> 6, m = i & 63;
        Ws[cc][m] = CW[((size_t)(hBase + m) * CCH + (c0 + cc)) * 3 + k];
      }
      for (int i = tid; i < 32 * 128; i += 256) {
        int cc = i >> 7, w = i & 127;
        const int ch = c0 + cc;
        float v = Hbuf[((size_t)(n * CCH + ch) * TTT + tt) * VVV + w];
        v = v * sc1[ch] + sh1[ch];          // BN1 (train-mode affine)
        Bs[cc][w] = v >= 0.0f ? v : at * v; // PReLU(a_tcn)
      }
      __syncthreads();
#pragma unroll
      for (int kk = 0; kk < 32; kk += 4) {
        const int mLoc = mTile * 16 + l16;
        v2f a;
        a.x = Ws[kk + klo][mLoc];
        a.y = Ws[kk + klo + 1][mLoc];
#pragma unroll
        for (int j = 0; j < 4; ++j) {
          const int col = (nGroup * 4 + j) * 16 + l16;
          v2f b;
          b.x = Bs[kk + klo][col];
          b.y = Bs[kk + klo + 1][col];
          acc[j] = wmma_k4(a, b, acc[j]);
        }
      }
      __syncthreads();
    }
  }

  const int mOff = (lane < 16) ? 0 : 8;
#pragma unroll
  for (int j = 0; j < 4; ++j) {
    const int w = (nGroup * 4 + j) * 16 + l16;
#pragma unroll
    for (int i = 0; i < 8; ++i) {
      const int h = hBase + mTile * 16 + mOff + i;
      OUT2[((size_t)(n * CCH + h) * TTT + t) * VVV + w] = acc[j][i] + CB[h];
    }
  }
}

// -------------------------------------------------------------------------
// K6: out = prelu( bn2(conv_out) + x, a_out )
// -------------------------------------------------------------------------
__global__ __launch_bounds__(256) void k6_out(const float* __restrict__ C2,
                                              const float* __restrict__ X,
                                              const float* __restrict__ sc2,
                                              const float* __restrict__ sh2,
                                              const float* __restrict__ aOut,
                                              float* __restrict__ OUT) {
  const size_t total = (size_t)NN * CCH * TTT * VVV;
  const size_t i = (size_t)blockIdx.x * 256 + threadIdx.x;
  if (i >= total) return;
  const int c = (int)((i / (TTT * VVV)) % CCH);
  const float v = C2[i] * sc2[c] + sh2[c] + X[i];
  const float a = aOut[0];
  OUT[i] = v >= 0.0f ? v : a * v;
}

// -------------------------------------------------------------------------
extern "C" void kernel_launch(void* const* d_in, const int* in_sizes, int n_in,
                              void* d_out, int out_size, void* d_ws,
                              size_t ws_size, hipStream_t stream) {
  const float* X  = (const float*)d_in[0];
  const float* A  = (const float*)d_in[1];
  const float* W0 = (const float*)d_in[2];
  const float* W1 = (const float*)d_in[3];
  const float* g1 = (const float*)d_in[4];
  const float* b1 = (const float*)d_in[5];
  const float* CW = (const float*)d_in[6];
  const float* CB = (const float*)d_in[7];
  const float* g2 = (const float*)d_in[8];
  const float* b2 = (const float*)d_in[9];
  const float* at = (const float*)d_in[10];
  const float* ao = (const float*)d_in[11];
  float* out = (float*)d_out;

  const size_t ETOT = (size_t)NN * CCH * TTT * VVV;  // 33,554,432
  float* ws = (float*)d_ws;
  float* XA  = ws;              // also reused for conv output
  float* Hb  = ws + ETOT;
  float* sc1 = ws + 2 * ETOT;
  float* sh1 = sc1 + CCH;
  float* sc2 = sh1 + CCH;
  float* sh2 = sc2 + CCH;

  dim3 blk(256);
  k1_xa<<<dim3(NN * TTT, CCH / 64), blk, 0, stream>>>(X, A, XA);
  k2_gcn<<<dim3(NN * TTT, CCH / 64), blk, 0, stream>>>(XA, W0, W1, X, Hb);
  k_bnstats<<<dim3(CCH), blk, 0, stream>>>(Hb, g1, b1, sc1, sh1);
  k4_conv<<<dim3(NN * TTT, CCH / 64), blk, 0, stream>>>(Hb, CW, CB, sc1, sh1,
                                                        at, XA);
  k_bnstats<<<dim3(CCH), blk, 0, stream>>>(XA, g2, b2, sc2, sh2);
  k6_out<<<dim3((unsigned)((ETOT + 255) / 256)), blk, 0, stream>>>(XA, X, sc2,
                                                                   sh2, ao, out);
}